// Network_66726611911212
// MI455X (gfx1250) — compile-verified
//
#include <hip/hip_runtime.h>
#include <math.h>

#define BB 2
#define NN 8192
#define KK 16
#define NEG_SLOPE 0.2f
#define EPS 1e-7f

typedef __attribute__((ext_vector_type(2))) float v2f;
typedef __attribute__((ext_vector_type(8))) float v8f;

// ---------------------------------------------------------------------------
// K0: per-point squared norms
// ---------------------------------------------------------------------------
__global__ __launch_bounds__(256) void xx_kernel(const float* __restrict__ xyz,
                                                 float* __restrict__ xx) {
  int i = blockIdx.x * 256 + threadIdx.x;
  if (i < BB * NN) {
    float x = xyz[i * 3 + 0], y = xyz[i * 3 + 1], z = xyz[i * 3 + 2];
    xx[i] = x * x + y * y + z * z;
  }
}

// ---------------------------------------------------------------------------
// K1: KNN via WMMA Gram tiles, 4 candidate tiles (64 candidates) per iter.
// One wave per 16 queries. v_wmma_f32_16x16x4_f32 computes a 16x16 tile of
// inner[candidate, query] (xyz padded to K=4). Ranking score =
// 2*inner - ||c||^2 (dropping the per-query constant). Each lane keeps a
// sorted top-16 list for query (lane&15) over its candidate half-rows;
// the two half-lane lists are merged through LDS at the end.
// ---------------------------------------------------------------------------
__global__ __launch_bounds__(32) void knn_kernel(const float* __restrict__ xyz,
                                                 const float* __restrict__ xx,
                                                 int* __restrict__ knn) {
  const int b = blockIdx.y;
  const int q0 = blockIdx.x * 16;
  const int lane = threadIdx.x;
  const int qcol = lane & 15;
  const bool hi = lane >= 16;
  const float* base = xyz + (size_t)b * NN * 3;

  // B operand: 4x16 query tile. K rows 0,1 in lanes 0-15, rows 2,3 in 16-31.
  v2f bop;
  {
    const float* qp = base + (size_t)(q0 + qcol) * 3;
    float qx = qp[0], qy = qp[1], qz = qp[2];  // single b96 load
    bop[0] = hi ? qz : qx;
    bop[1] = hi ? 0.0f : qy;
  }
  const float* xxb = xx + (size_t)b * NN;
  const int mbase = hi ? 8 : 0;

  float td[KK];
  int ti[KK];
#pragma unroll
  for (int i = 0; i < KK; ++i) { td[i] = -3.4e38f; ti[i] = 0; }

  for (int c0 = 0; c0 < NN; c0 += 64) {
    if (c0 + 64 < NN)
      __builtin_prefetch(base + (size_t)(c0 + 64 + qcol) * 3, 0, 3);

    // Batch all loads for 4 tiles first (4x b96 + 8x b128), then 4 WMMAs.
    v2f aop[4];
    float4 xlo[4], xhi[4];
#pragma unroll
    for (int u = 0; u < 4; ++u) {
      const int cc = c0 + u * 16;
      const float* ap = base + (size_t)(cc + qcol) * 3;
      float ax = ap[0], ay = ap[1], az = ap[2];  // one b96 load, no branch
      aop[u][0] = hi ? az : ax;
      aop[u][1] = hi ? 0.0f : ay;
      xlo[u] = *(const float4*)(xxb + cc + mbase);
      xhi[u] = *(const float4*)(xxb + cc + mbase + 4);
    }
    v8f acc[4];
#pragma unroll
    for (int u = 0; u < 4; ++u) {
      v8f z = {0.f, 0.f, 0.f, 0.f, 0.f, 0.f, 0.f, 0.f};
      acc[u] = __builtin_amdgcn_wmma_f32_16x16x4_f32(false, aop[u], false, bop,
                                                     (short)0, z, false, false);
    }
#pragma unroll
    for (int u = 0; u < 4; ++u) {
      const int cbase = c0 + u * 16 + mbase;
      float xs[8] = {xlo[u].x, xlo[u].y, xlo[u].z, xlo[u].w,
                     xhi[u].x, xhi[u].y, xhi[u].z, xhi[u].w};
#pragma unroll
      for (int r = 0; r < 8; ++r) {
        const float v = 2.0f * acc[u][r] - xs[r];
        if (v > td[KK - 1]) {  // threshold check; rare sorted insertion
          td[KK - 1] = v; ti[KK - 1] = cbase + r;
#pragma unroll
          for (int i = KK - 1; i > 0; --i) {
            if (td[i] > td[i - 1]) {
              float tv = td[i]; td[i] = td[i - 1]; td[i - 1] = tv;
              int tj = ti[i]; ti[i] = ti[i - 1]; ti[i - 1] = tj;
            }
          }
        }
      }
    }
  }

  // merge the two sorted half-lane lists per query
  __shared__ float sd[32 * KK];
  __shared__ int si[32 * KK];
#pragma unroll
  for (int i = 0; i < KK; ++i) { sd[lane * KK + i] = td[i]; si[lane * KK + i] = ti[i]; }
  __syncthreads();
  if (lane < 16) {
    int i = 0, j = 0;
    int* outp = knn + ((size_t)b * NN + (q0 + lane)) * KK;
#pragma unroll
    for (int t = 0; t < KK; ++t) {
      float va = sd[lane * KK + i];
      float vb = sd[(lane + 16) * KK + j];
      if (va >= vb) { outp[t] = si[lane * KK + i]; ++i; }
      else          { outp[t] = si[(lane + 16) * KK + j]; ++j; }
    }
  }
}

// ---------------------------------------------------------------------------
// K2: relative-pos encoding (c=4) -> w1_lin (16x4) -> VN leaky relu (w1_dir 16x16)
// One thread per (b,n,k).
// ---------------------------------------------------------------------------
__global__ __launch_bounds__(256) void enc1_kernel(const float* __restrict__ xyz,
                                                   const int* __restrict__ knn,
                                                   const float* __restrict__ w_lin,
                                                   const float* __restrict__ w_dir,
                                                   float* __restrict__ fxyz) {
  __shared__ float sLin[64];
  __shared__ float sDir[256];
  if (threadIdx.x < 64) sLin[threadIdx.x] = w_lin[threadIdx.x];
  sDir[threadIdx.x] = w_dir[threadIdx.x];
  __syncthreads();
  const int e = blockIdx.x * 256 + threadIdx.x;  // (b*N + n)*K + k
  if (e >= BB * NN * KK) return;
  const int bn = e / KK;
  const int b = bn / NN;
  const int nb = knn[e];
  const float* pc = xyz + (size_t)bn * 3;
  const float* pn = xyz + ((size_t)b * NN + nb) * 3;
  float cx = pc[0], cy = pc[1], cz = pc[2];
  float nx = pn[0], ny = pn[1], nz = pn[2];
  float rx = cx - nx, ry = cy - ny, rz = cz - nz;
  float dist = sqrtf(rx * rx + ry * ry + rz * rz);
  float enc[4][3] = {{dist, dist, dist}, {rx, ry, rz}, {cx, cy, cz}, {nx, ny, nz}};
  float x1[48], dv[48];
#pragma unroll
  for (int o = 0; o < 16; ++o)
#pragma unroll
    for (int t = 0; t < 3; ++t) {
      float a = 0.f;
#pragma unroll
      for (int c = 0; c < 4; ++c) a += sLin[o * 4 + c] * enc[c][t];
      x1[o * 3 + t] = a;
    }
#pragma unroll
  for (int o = 0; o < 16; ++o)
#pragma unroll
    for (int t = 0; t < 3; ++t) {
      float a = 0.f;
#pragma unroll
      for (int c = 0; c < 16; ++c) a += sDir[o * 16 + c] * x1[c * 3 + t];
      dv[o * 3 + t] = a;
    }
  float dot[3] = {0, 0, 0}, dns[3] = {0, 0, 0};
#pragma unroll
  for (int c = 0; c < 16; ++c)
#pragma unroll
    for (int t = 0; t < 3; ++t) {
      dot[t] += x1[c * 3 + t] * dv[c * 3 + t];
      dns[t] += dv[c * 3 + t] * dv[c * 3 + t];
    }
  float* outp = fxyz + (size_t)e * 48;
#pragma unroll
  for (int c = 0; c < 16; ++c)
#pragma unroll
    for (int t = 0; t < 3; ++t) {
      float x = x1[c * 3 + t], d = dv[c * 3 + t];
      float alt = x - dot[t] / (dns[t] + EPS) * d;
      float r = (dot[t] >= 0.f) ? x : alt;
      outp[c * 3 + t] = NEG_SLOPE * x + (1.0f - NEG_SLOPE) * r;
    }
}

// ---------------------------------------------------------------------------
// K4: VN block c=16 -> 16 (w2_lin) + leaky (w2_dir), in-place on fxyz.
// ---------------------------------------------------------------------------
__global__ __launch_bounds__(256) void vnblock16_kernel(const float* __restrict__ w_lin,
                                                        const float* __restrict__ w_dir,
                                                        float* __restrict__ fxyz) {
  __shared__ float sLin[256];
  __shared__ float sDir[256];
  sLin[threadIdx.x] = w_lin[threadIdx.x];
  sDir[threadIdx.x] = w_dir[threadIdx.x];
  __syncthreads();
  const int e = blockIdx.x * 256 + threadIdx.x;
  if (e >= BB * NN * KK) return;
  float* p = fxyz + (size_t)e * 48;
  float x0[48], x1[48], dv[48];
#pragma unroll
  for (int i = 0; i < 48; ++i) x0[i] = p[i];
#pragma unroll
  for (int o = 0; o < 16; ++o)
#pragma unroll
    for (int t = 0; t < 3; ++t) {
      float a = 0.f;
#pragma unroll
      for (int c = 0; c < 16; ++c) a += sLin[o * 16 + c] * x0[c * 3 + t];
      x1[o * 3 + t] = a;
    }
#pragma unroll
  for (int o = 0; o < 16; ++o)
#pragma unroll
    for (int t = 0; t < 3; ++t) {
      float a = 0.f;
#pragma unroll
      for (int c = 0; c < 16; ++c) a += sDir[o * 16 + c] * x1[c * 3 + t];
      dv[o * 3 + t] = a;
    }
  float dot[3] = {0, 0, 0}, dns[3] = {0, 0, 0};
#pragma unroll
  for (int c = 0; c < 16; ++c)
#pragma unroll
    for (int t = 0; t < 3; ++t) {
      dot[t] += x1[c * 3 + t] * dv[c * 3 + t];
      dns[t] += dv[c * 3 + t] * dv[c * 3 + t];
    }
#pragma unroll
  for (int c = 0; c < 16; ++c)
#pragma unroll
    for (int t = 0; t < 3; ++t) {
      float x = x1[c * 3 + t], d = dv[c * 3 + t];
      float alt = x - dot[t] / (dns[t] + EPS) * d;
      float r = (dot[t] >= 0.f) ? x : alt;
      p[c * 3 + t] = NEG_SLOPE * x + (1.0f - NEG_SLOPE) * r;
    }
}

// ---------------------------------------------------------------------------
// K3: attention pooling #1. One wave per point; lane == channel.
// fs = concat(gather(feature), fxyz) (K x 32 x 3) staged in LDS.
// ---------------------------------------------------------------------------
__global__ __launch_bounds__(32) void attpool1_kernel(const float* __restrict__ feature,
                                                      const float* __restrict__ fxyz,
                                                      const float* __restrict__ fc,
                                                      const float* __restrict__ mlp,
                                                      const float* __restrict__ dir,
                                                      const int* __restrict__ knn,
                                                      float* __restrict__ fagg) {
  const int e = blockIdx.x;  // b*N + n
  const int b = e / NN;
  const int lane = threadIdx.x;
  __shared__ float sFS[KK * 96];
  __shared__ float sW[1024];
  __shared__ float sAgg[96];
  __shared__ float sY[48];
  __shared__ float sD[48];
  __shared__ int sNb[KK];
  if (lane < KK) sNb[lane] = knn[e * KK + lane];
  for (int i = lane; i < 1024; i += 32) sW[i] = fc[i];
  __syncthreads();
  for (int i = lane; i < KK * 96; i += 32) {
    int k = i / 96, r = i % 96, c = r / 3, t = r % 3;
    float v;
    if (c < 16) {
      v = feature[(((size_t)b * NN + sNb[k]) * 16 + c) * 3 + t];
    } else {
      v = fxyz[((size_t)e * KK + k) * 48 + (c - 16) * 3 + t];
    }
    sFS[i] = v;
  }
  __syncthreads();
  const int o = lane;
  float w[32];
#pragma unroll
  for (int c = 0; c < 32; ++c) w[c] = sW[o * 32 + c];
#pragma unroll
  for (int t = 0; t < 3; ++t) {
    float att[KK];
    float m = -3.4e38f;
#pragma unroll
    for (int k = 0; k < KK; ++k) {
      float a = 0.f;
#pragma unroll
      for (int c = 0; c < 32; ++c) a += w[c] * sFS[k * 96 + c * 3 + t];
      att[k] = a;
      m = fmaxf(m, a);
    }
    float s = 0.f;
#pragma unroll
    for (int k = 0; k < KK; ++k) { att[k] = __expf(att[k] - m); s += att[k]; }
    float inv = 1.f / s;
    float a = 0.f;
#pragma unroll
    for (int k = 0; k < KK; ++k) a += sFS[k * 96 + o * 3 + t] * att[k];
    sAgg[o * 3 + t] = a * inv;
  }
  __syncthreads();
  if (o < 16) {
#pragma unroll
    for (int t = 0; t < 3; ++t) {
      float y = 0.f;
#pragma unroll
      for (int c = 0; c < 32; ++c) y += mlp[o * 32 + c] * sAgg[c * 3 + t];
      sY[o * 3 + t] = y;
    }
  }
  __syncthreads();
  if (o < 16) {
#pragma unroll
    for (int t = 0; t < 3; ++t) {
      float dd = 0.f;
#pragma unroll
      for (int c = 0; c < 16; ++c) dd += dir[o * 16 + c] * sY[c * 3 + t];
      sD[o * 3 + t] = dd;
    }
  }
  __syncthreads();
  if (o < 16) {
#pragma unroll
    for (int t = 0; t < 3; ++t) {
      float dot = 0.f, dns = 0.f;
#pragma unroll
      for (int c = 0; c < 16; ++c) {
        dot += sY[c * 3 + t] * sD[c * 3 + t];
        dns += sD[c * 3 + t] * sD[c * 3 + t];
      }
      float x = sY[o * 3 + t], d = sD[o * 3 + t];
      float alt = x - dot / (dns + EPS) * d;
      float r = (dot >= 0.f) ? x : alt;
      fagg[((size_t)e * 16 + o) * 3 + t] = NEG_SLOPE * x + (1.f - NEG_SLOPE) * r;
    }
  }
}

// ---------------------------------------------------------------------------
// K5: attention pooling #2 (c=32 everywhere), writes final output (B,N,32,3).
// ---------------------------------------------------------------------------
__global__ __launch_bounds__(32) void attpool2_kernel(const float* __restrict__ fagg,
                                                      const float* __restrict__ fxyz,
                                                      const float* __restrict__ fc,
                                                      const float* __restrict__ mlp,
                                                      const float* __restrict__ dir,
                                                      const int* __restrict__ knn,
                                                      float* __restrict__ out) {
  const int e = blockIdx.x;
  const int b = e / NN;
  const int lane = threadIdx.x;
  __shared__ float sFS[KK * 96];
  __shared__ float sW[1024];
  __shared__ float sAgg[96];
  __shared__ float sY[96];
  __shared__ float sD[96];
  __shared__ int sNb[KK];
  if (lane < KK) sNb[lane] = knn[e * KK + lane];
  for (int i = lane; i < 1024; i += 32) sW[i] = fc[i];
  __syncthreads();
  for (int i = lane; i < KK * 96; i += 32) {
    int k = i / 96, r = i % 96, c = r / 3, t = r % 3;
    float v;
    if (c < 16) {
      v = fagg[(((size_t)b * NN + sNb[k]) * 16 + c) * 3 + t];
    } else {
      v = fxyz[((size_t)e * KK + k) * 48 + (c - 16) * 3 + t];
    }
    sFS[i] = v;
  }
  __syncthreads();
  const int o = lane;
  float w[32];
#pragma unroll
  for (int c = 0; c < 32; ++c) w[c] = sW[o * 32 + c];
#pragma unroll
  for (int t = 0; t < 3; ++t) {
    float att[KK];
    float m = -3.4e38f;
#pragma unroll
    for (int k = 0; k < KK; ++k) {
      float a = 0.f;
#pragma unroll
      for (int c = 0; c < 32; ++c) a += w[c] * sFS[k * 96 + c * 3 + t];
      att[k] = a;
      m = fmaxf(m, a);
    }
    float s = 0.f;
#pragma unroll
    for (int k = 0; k < KK; ++k) { att[k] = __expf(att[k] - m); s += att[k]; }
    float inv = 1.f / s;
    float a = 0.f;
#pragma unroll
    for (int k = 0; k < KK; ++k) a += sFS[k * 96 + o * 3 + t] * att[k];
    sAgg[o * 3 + t] = a * inv;
  }
  __syncthreads();
#pragma unroll
  for (int t = 0; t < 3; ++t) {
    float y = 0.f;
#pragma unroll
    for (int c = 0; c < 32; ++c) y += mlp[o * 32 + c] * sAgg[c * 3 + t];
    sY[o * 3 + t] = y;
  }
  __syncthreads();
#pragma unroll
  for (int t = 0; t < 3; ++t) {
    float dd = 0.f;
#pragma unroll
    for (int c = 0; c < 32; ++c) dd += dir[o * 32 + c] * sY[c * 3 + t];
    sD[o * 3 + t] = dd;
  }
  __syncthreads();
#pragma unroll
  for (int t = 0; t < 3; ++t) {
    float dot = 0.f, dns = 0.f;
#pragma unroll
    for (int c = 0; c < 32; ++c) {
      dot += sY[c * 3 + t] * sD[c * 3 + t];
      dns += sD[c * 3 + t] * sD[c * 3 + t];
    }
    float x = sY[o * 3 + t], d = sD[o * 3 + t];
    float alt = x - dot / (dns + EPS) * d;
    float r = (dot >= 0.f) ? x : alt;
    out[((size_t)e * 32 + o) * 3 + t] = NEG_SLOPE * x + (1.f - NEG_SLOPE) * r;
  }
}

// ---------------------------------------------------------------------------
extern "C" void kernel_launch(void* const* d_in, const int* in_sizes, int n_in,
                              void* d_out, int out_size, void* d_ws, size_t ws_size,
                              hipStream_t stream) {
  const float* xyz = (const float*)d_in[0];
  const float* feature = (const float*)d_in[1];
  const float* w1_lin = (const float*)d_in[2];
  const float* w1_dir = (const float*)d_in[3];
  const float* ap1_fc = (const float*)d_in[4];
  const float* ap1_mlp = (const float*)d_in[5];
  const float* ap1_dir = (const float*)d_in[6];
  const float* w2_lin = (const float*)d_in[7];
  const float* w2_dir = (const float*)d_in[8];
  const float* ap2_fc = (const float*)d_in[9];
  const float* ap2_mlp = (const float*)d_in[10];
  const float* ap2_dir = (const float*)d_in[11];
  float* out = (float*)d_out;

  char* ws = (char*)d_ws;
  float* xx = (float*)ws;                                        // 16384 f
  int* knn = (int*)(ws + 65536);                                 // 262144 i
  float* fxyz = (float*)(ws + 65536 + 1048576);                  // B*N*K*48 f
  float* fagg = (float*)(ws + 65536 + 1048576 +
                         (size_t)BB * NN * KK * 48 * sizeof(float));  // B*N*16*3 f

  xx_kernel<<<dim3((BB * NN + 255) / 256), dim3(256), 0, stream>>>(xyz, xx);
  knn_kernel<<<dim3(NN / 16, BB), dim3(32), 0, stream>>>(xyz, xx, knn);
  enc1_kernel<<<dim3(BB * NN * KK / 256), dim3(256), 0, stream>>>(xyz, knn, w1_lin, w1_dir, fxyz);
  attpool1_kernel<<<dim3(BB * NN), dim3(32), 0, stream>>>(feature, fxyz, ap1_fc, ap1_mlp, ap1_dir, knn, fagg);
  vnblock16_kernel<<<dim3(BB * NN * KK / 256), dim3(256), 0, stream>>>(w2_lin, w2_dir, fxyz);
  attpool2_kernel<<<dim3(BB * NN), dim3(32), 0, stream>>>(fagg, fxyz, ap2_fc, ap2_mlp, ap2_dir, knn, out);
}